// SqueezeExcitation_79139067396494
// MI455X (gfx1250) — compile-verified
//
#include <hip/hip_runtime.h>
#include <hip/hip_bf16.h>
#include <math.h>

// Problem constants (from reference setup_inputs)
#define B_   32
#define C_   768
#define CR_  192
#define HW_  1024           // 32*32
#define K1_  C_             // GEMM1 K
#define K2_  CR_            // GEMM2 K

typedef __attribute__((ext_vector_type(2))) float v2f;
typedef __attribute__((ext_vector_type(8))) float v8f;

__device__ __forceinline__ float sigmoidf_fast(float v) {
    return 1.0f / (1.0f + __expf(-v));
}

// ---------------------------------------------------------------------------
// Kernel 1: squeeze — global average pool over H*W for each (b,c).
// One 256-thread block per (b,c); each thread loads one float4 (256*4 = 1024).
// ---------------------------------------------------------------------------
__global__ __launch_bounds__(256) void se_squeeze(const float* __restrict__ x,
                                                  float* __restrict__ s) {
    const int bc = blockIdx.x;                       // 0 .. B*C-1
    const float4* p = (const float4*)(x + (size_t)bc * HW_);
    float4 v = p[threadIdx.x];
    float sum = v.x + v.y + v.z + v.w;

    // wave32 reduction
    #pragma unroll
    for (int o = 16; o > 0; o >>= 1)
        sum += __shfl_down(sum, o, 32);

    __shared__ float tmp[8];
    const int lane = threadIdx.x & 31;
    const int wave = threadIdx.x >> 5;
    if (lane == 0) tmp[wave] = sum;
    __syncthreads();
    if (threadIdx.x < 8) {
        float t = tmp[threadIdx.x];
        #pragma unroll
        for (int o = 4; o > 0; o >>= 1)
            t += __shfl_down(t, o, 8);
        if (threadIdx.x == 0) s[bc] = t * (1.0f / (float)HW_);
    }
}

// ---------------------------------------------------------------------------
// Kernel 2: excitation MLP using V_WMMA_F32_16X16X4_F32.
// Single workgroup, 8 waves. Each wave computes 16x16 f32 tiles.
//   h = swish(s @ w1^T + b1)   [32 x 192]
//   g = sigmoid(h @ w2^T + b2) [32 x 768]
// A layout (16x4 f32): lanes 0-15 hold K=0,1 ; lanes 16-31 hold K=2,3 (per M row)
// B layout (4x16 f32): lanes 0-15 hold K=0,1 ; lanes 16-31 hold K=2,3 (per N col)
// C/D layout: VGPR r -> M = r + 8*(lane>=16), N = lane&15
// ---------------------------------------------------------------------------
__global__ __launch_bounds__(256) void se_mlp_wmma(const float* __restrict__ s,
                                                   const float* __restrict__ w1,
                                                   const float* __restrict__ b1,
                                                   const float* __restrict__ w2,
                                                   const float* __restrict__ b2,
                                                   float* __restrict__ h,
                                                   float* __restrict__ g) {
    const int wave = threadIdx.x >> 5;
    const int lane = threadIdx.x & 31;
    const int half = lane >> 4;        // 0: K pair {0,1}, 1: K pair {2,3}
    const int l    = lane & 15;        // row (A) / col (B) / N (D)
    const int koff = 2 * half;

    // ---------------- GEMM1: h = swish(s[32,768] @ w1[192,768]^T + b1) -----
    // tiles: 2 (M) x 12 (N) = 24 ; 8 waves -> 3 tiles each (uniform, no divergence)
    for (int t = wave; t < 24; t += 8) {
        const int mt = t & 1;
        const int nt = t >> 1;
        const int m0 = mt * 16;
        const int n0 = nt * 16;
        const float* arow = s  + (size_t)(m0 + l) * K1_ + koff;
        const float* brow = w1 + (size_t)(n0 + l) * K1_ + koff;

        v8f acc = {};
        #pragma unroll 4
        for (int k = 0; k < K1_; k += 4) {
            v2f a = *(const v2f*)(arow + k);
            v2f b = *(const v2f*)(brow + k);
            acc = __builtin_amdgcn_wmma_f32_16x16x4_f32(
                false, a, false, b, (short)0, acc, false, false);
        }
        const float bias = b1[n0 + l];
        #pragma unroll
        for (int r = 0; r < 8; ++r) {
            const int m = m0 + r + 8 * half;
            float v = acc[r] + bias;
            v = v * sigmoidf_fast(v);            // swish
            h[(size_t)m * CR_ + n0 + l] = v;
        }
    }

    __syncthreads();   // all of h must be visible before GEMM2

    // ---------------- GEMM2: g = sigmoid(h[32,192] @ w2[768,192]^T + b2) ---
    // tiles: 2 (M) x 48 (N) = 96 ; 8 waves -> 12 tiles each (uniform)
    for (int t = wave; t < 96; t += 8) {
        const int mt = t & 1;
        const int nt = t >> 1;
        const int m0 = mt * 16;
        const int n0 = nt * 16;
        const float* arow = h  + (size_t)(m0 + l) * K2_ + koff;
        const float* brow = w2 + (size_t)(n0 + l) * K2_ + koff;

        v8f acc = {};
        #pragma unroll 4
        for (int k = 0; k < K2_; k += 4) {
            v2f a = *(const v2f*)(arow + k);
            v2f b = *(const v2f*)(brow + k);
            acc = __builtin_amdgcn_wmma_f32_16x16x4_f32(
                false, a, false, b, (short)0, acc, false, false);
        }
        const float bias = b2[n0 + l];
        #pragma unroll
        for (int r = 0; r < 8; ++r) {
            const int m = m0 + r + 8 * half;
            float v = acc[r] + bias;
            g[(size_t)m * C_ + n0 + l] = sigmoidf_fast(v);
        }
    }
}

// ---------------------------------------------------------------------------
// Kernel 3: scale — out = x * g[b,c] (broadcast over HW). float4 streaming.
// ---------------------------------------------------------------------------
__global__ __launch_bounds__(256) void se_scale(const float* __restrict__ x,
                                                const float* __restrict__ g,
                                                float* __restrict__ out) {
    const size_t idx4 = (size_t)blockIdx.x * blockDim.x + threadIdx.x;
    const size_t total4 = (size_t)B_ * C_ * HW_ / 4;
    if (idx4 >= total4) return;
    const size_t bc = idx4 >> 8;                 // HW_/4 = 256 float4 per (b,c)
    const float gv = g[bc];
    float4 v = ((const float4*)x)[idx4];
    v.x *= gv; v.y *= gv; v.z *= gv; v.w *= gv;
    ((float4*)out)[idx4] = v;
}

extern "C" void kernel_launch(void* const* d_in, const int* in_sizes, int n_in,
                              void* d_out, int out_size, void* d_ws, size_t ws_size,
                              hipStream_t stream) {
    const float* x  = (const float*)d_in[0];   // [32,768,32,32]
    const float* w1 = (const float*)d_in[1];   // [192,768]
    const float* b1 = (const float*)d_in[2];   // [192]
    const float* w2 = (const float*)d_in[3];   // [768,192]
    const float* b2 = (const float*)d_in[4];   // [768]
    float* out = (float*)d_out;                // [32,768,32,32]

    // workspace layout
    float* s = (float*)d_ws;                               // 32*768   = 24576 f
    float* h = s + (size_t)B_ * C_;                        // 32*192   =  6144 f
    float* g = h + (size_t)B_ * CR_;                       // 32*768   = 24576 f

    // 1) squeeze: one block per (b,c)
    se_squeeze<<<B_ * C_, 256, 0, stream>>>(x, s);

    // 2) excitation MLP: single workgroup, WMMA f32 16x16x4
    se_mlp_wmma<<<1, 256, 0, stream>>>(s, w1, b1, w2, b2, h, g);

    // 3) scale: 32*768*1024/4 float4s / 256 threads
    const int nblocks = (int)(((size_t)B_ * C_ * HW_ / 4 + 255) / 256);
    se_scale<<<nblocks, 256, 0, stream>>>(x, g, out);
}